// ProposalGeneration_56118042689952
// MI455X (gfx1250) — compile-verified
//
#include <hip/hip_runtime.h>

typedef __attribute__((ext_vector_type(2))) float v2f;
typedef __attribute__((ext_vector_type(8))) float v8f;
typedef __attribute__((ext_vector_type(4))) unsigned int v4u;
typedef __attribute__((ext_vector_type(8))) int v8i;
typedef __attribute__((ext_vector_type(4))) int v4i;

#define T_DIM 256
#define D_DIM 512
#define L_DIM 32
#define C_DIM 8
#define M_DIM (L_DIM * L_DIM * C_DIM) /* 8192 */
#define KC 32
#define LDA 36  /* 32 DW row + 4 DW pad: 16 distinct banks for A column reads */
#define LDB 80  /* 64 DW row + 16 DW pad: +32-bank shift between lane halves */

#define HAVE_TDM (__has_builtin(__builtin_amdgcn_tensor_load_to_lds) && \
                  __has_builtin(__builtin_amdgcn_s_wait_tensorcnt))

// D# group1 static descriptors (ISA 8.4):
//  w0: data_size=2(4B)<<16 | pad_enable<<20 | pad_interval<<22 | pad_amount<<25
//  w1[31:16]=tensor_dim0.lo16  w2[31:16]=tensor_dim1.lo16
//  w3[31:16]=tile_dim0         w4[15:0]=tile_dim1
//  w5=tensor_dim0_stride[31:0]
// A tile: 64 rows x 32 elems, row stride 256, pad 32DW+4DW -> LDS stride 36
#define G1_A_W0 (0x20000u | (1u << 20) | (4u << 22) | (3u << 25))
// B tile: 32 rows x 64 elems, row stride 512, pad 64DW+16DW -> LDS stride 80
#define G1_B_W0 (0x20000u | (1u << 20) | (5u << 22) | (15u << 25))

// GEMM fc[b] = Wc[8192x256] * f[b][256x512], fused fm (clip mean over c).
// Double-buffered TDM staging: wave 0 issues next chunk's descriptors while
// all 8 waves run WMMA on the current buffer; TENSORcnt + barrier ordering.
__global__ __launch_bounds__(256) void fc_gemm_kernel(
    const float* __restrict__ f,   // [B, T, D]
    const float* __restrict__ Wc,  // [M, T] (= [L,L,C,T] flattened)
    float* __restrict__ fc,        // [B, M, D]
    float* __restrict__ fm)        // [B, L*L, D]
{
    __shared__ float As[2][64 * LDA];
    __shared__ float Bs[2][KC * LDB];

    const int n0 = blockIdx.x * 64;
    const int m0 = blockIdx.y * 64;
    const int b  = blockIdx.z;

    const int tid  = threadIdx.x;
    const int wave = tid >> 5;
    const int lane = tid & 31;
    const int lm   = lane & 15;  // N (or M) index within 16
    const int lh   = lane >> 4;  // lane half selects K pair / M group
    const int wm   = wave & 3;   // wave M sub-tile (16 rows)
    const int wn   = wave >> 2;  // wave N sub-tile (32 cols)

    // K-range clamping from Wc sparsity: tile has constant window-start i,
    // j in [j0, j0+8); support is [i*8, (j0+8)*8); all-zero tiles skip.
    const int ij0 = blockIdx.y * 8;
    const int i_w = ij0 >> 5;
    const int j0  = ij0 & 31;
    int kmin = 0, kmax = 0;
    if (j0 + 7 >= i_w) {
        kmin = (i_w << 3) & ~(KC - 1);
        kmax = (j0 + 8) << 3;
    }
    const int nch = (kmax - kmin) / KC;

    const float* fB = f + (size_t)b * T_DIM * D_DIM;

    v8f acc0 = {};
    v8f acc1 = {};

#if HAVE_TDM
    const v8i g1A = { (int)G1_A_W0, 32 << 16, 64 << 16, 32 << 16, 64, 256, 0, 0 };
    const v8i g1B = { (int)G1_B_W0, 64 << 16, 32 << 16, 64 << 16, 32, 512, 0, 0 };
    const v4i zero4 = {};
    const v8i zero8 = {};

    auto issue = [&](int sel, int k0) {
        unsigned long long gA =
            (unsigned long long)(size_t)(Wc + (size_t)m0 * T_DIM + k0);
        unsigned long long gB =
            (unsigned long long)(size_t)(fB + (size_t)k0 * D_DIM + n0);
        v4u g0A = { 1u, (unsigned)(size_t)(void*)&As[sel][0],
                    (unsigned)gA, (unsigned)(gA >> 32) | 0x80000000u };
        v4u g0B = { 1u, (unsigned)(size_t)(void*)&Bs[sel][0],
                    (unsigned)gB, (unsigned)(gB >> 32) | 0x80000000u };
        __builtin_amdgcn_tensor_load_to_lds(g0A, g1A, zero4, zero4, zero8, 0);
        __builtin_amdgcn_tensor_load_to_lds(g0B, g1B, zero4, zero4, zero8, 0);
    };

    if (nch > 0 && wave == 0) issue(0, kmin);

    for (int c = 0; c < nch; ++c) {
        if (wave == 0) __builtin_amdgcn_s_wait_tensorcnt((unsigned short)0);
        __syncthreads();  // buffer c ready; buffer c^1 drained (compute c-1 done)
        if (wave == 0 && c + 1 < nch) issue((c + 1) & 1, kmin + (c + 1) * KC);

        const float* Arow = &As[c & 1][(wm * 16 + lm) * LDA];
        const float* Bb   = &Bs[c & 1][0];
#pragma unroll
        for (int kk = 0; kk < KC; kk += 4) {
            // 16x4 f32 A layout: lanes 0-15 K={0,1}, lanes 16-31 K={2,3}
            v2f a;
            a.x = Arow[kk + 2 * lh + 0];
            a.y = Arow[kk + 2 * lh + 1];
            // 4x16 f32 B layout: lane = N, lane half selects K pair
            const float* Bp = &Bb[(kk + 2 * lh) * LDB + wn * 32 + lm];
            v2f b0; b0.x = Bp[0];  b0.y = Bp[LDB];
            v2f b1; b1.x = Bp[16]; b1.y = Bp[LDB + 16];
            acc0 = __builtin_amdgcn_wmma_f32_16x16x4_f32(
                false, a, false, b0, (short)0, acc0, false, false);
            acc1 = __builtin_amdgcn_wmma_f32_16x16x4_f32(
                false, a, false, b1, (short)0, acc1, false, false);
        }
    }
#else
    for (int c = 0; c < nch; ++c) {
        int k0 = kmin + c * KC;
        __syncthreads();
        {   // manual staging fallback (single buffer)
            int row = tid >> 3;
            int c4  = (tid & 7) << 2;
            *reinterpret_cast<float4*>(&As[0][row * LDA + c4]) =
                *reinterpret_cast<const float4*>(Wc + (size_t)(m0 + row) * T_DIM + k0 + c4);
            *reinterpret_cast<float4*>(&As[0][(row + 32) * LDA + c4]) =
                *reinterpret_cast<const float4*>(Wc + (size_t)(m0 + row + 32) * T_DIM + k0 + c4);
            int rowb = tid >> 4;
            int c4b  = (tid & 15) << 2;
            *reinterpret_cast<float4*>(&Bs[0][rowb * LDB + c4b]) =
                *reinterpret_cast<const float4*>(fB + (size_t)(k0 + rowb) * D_DIM + n0 + c4b);
            *reinterpret_cast<float4*>(&Bs[0][(rowb + 16) * LDB + c4b]) =
                *reinterpret_cast<const float4*>(fB + (size_t)(k0 + rowb + 16) * D_DIM + n0 + c4b);
        }
        __syncthreads();
        const float* Arow = &As[0][(wm * 16 + lm) * LDA];
#pragma unroll
        for (int kk = 0; kk < KC; kk += 4) {
            v2f a;
            a.x = Arow[kk + 2 * lh + 0];
            a.y = Arow[kk + 2 * lh + 1];
            const float* Bp = &Bs[0][(kk + 2 * lh) * LDB + wn * 32 + lm];
            v2f b0; b0.x = Bp[0];  b0.y = Bp[LDB];
            v2f b1; b1.x = Bp[16]; b1.y = Bp[LDB + 16];
            acc0 = __builtin_amdgcn_wmma_f32_16x16x4_f32(
                false, a, false, b0, (short)0, acc0, false, false);
            acc1 = __builtin_amdgcn_wmma_f32_16x16x4_f32(
                false, a, false, b1, (short)0, acc1, false, false);
        }
    }
#endif

    // D layout: VGPR r holds M = r + 8*lh, N = lm (+16 for acc1).
    // fc is written once and never re-read: stream it past the caches (NT).
    float* out = fc + ((size_t)b * M_DIM + (m0 + wm * 16 + 8 * lh)) * D_DIM
                    + (n0 + wn * 32 + lm);
    float s0 = 0.0f, s1 = 0.0f;
#pragma unroll
    for (int r = 0; r < 8; ++r) {
        __builtin_nontemporal_store(acc0[r], &out[(size_t)r * D_DIM]);
        __builtin_nontemporal_store(acc1[r], &out[(size_t)r * D_DIM + 16]);
        s0 += acc0[r];
        s1 += acc1[r];
    }
    // Fused fm: each lane-half holds one full c-group of 8 rows.
    int g = ((m0 + wm * 16) >> 3) + lh;
    float* fmp = fm + ((size_t)b * (L_DIM * L_DIM) + g) * D_DIM
                    + (n0 + wn * 32 + lm);
    fmp[0]  = s0 * 0.125f;
    fmp[16] = s1 * 0.125f;
}

// fb: AvgPool1d over time, kernel = stride = T/L = 8. One thread per output.
__global__ __launch_bounds__(256) void fb_kernel(
    const float* __restrict__ f, float* __restrict__ fb, int n)
{
    int idx = blockIdx.x * blockDim.x + threadIdx.x;
    if (idx >= n) return;
    int d  = idx % D_DIM;
    int bl = idx / D_DIM;  // (b*L + l)*8 == b*T + l*8 since L*8 == T
    const float* p = f + (size_t)bl * 8 * D_DIM + d;
    float s = 0.0f;
#pragma unroll
    for (int kk = 0; kk < 8; ++kk) s += p[(size_t)kk * D_DIM];
    fb[idx] = s * 0.125f;
}

extern "C" void kernel_launch(void* const* d_in, const int* in_sizes, int n_in,
                              void* d_out, int out_size, void* d_ws, size_t ws_size,
                              hipStream_t stream) {
    const float* f  = (const float*)d_in[0];   // [B, T, D]
    const float* Wc = (const float*)d_in[1];   // [L, L, C, T]
    const int B = in_sizes[0] / (T_DIM * D_DIM);

    float* fc = (float*)d_out;                           // [B, 8192, 512]
    float* fm = fc + (size_t)B * M_DIM * D_DIM;          // [B, 1024, 512]
    float* fb = fm + (size_t)B * L_DIM * L_DIM * D_DIM;  // [B, 32, 512]

    dim3 grid(D_DIM / 64, M_DIM / 64, B);
    fc_gemm_kernel<<<grid, 256, 0, stream>>>(f, Wc, fc, fm);

    int nfb = B * L_DIM * D_DIM;
    fb_kernel<<<(nfb + 255) / 256, 256, 0, stream>>>(f, fb, nfb);
}